// PhysicsInformedNN_73504070304394
// MI455X (gfx1250) — compile-verified
//
#include <hip/hip_runtime.h>
#include <hip/hip_bf16.h>

// CDNA5 / gfx1250 PINN residual kernel.
// Strategy: 13-component forward jet (value; d/dx,dy,dt; 5 second derivs;
// 4 third derivs) through the 8-layer tanh MLP. All jet components share the
// layer weight matrix, so each layer is a (16 pts x din) x (din x dout) GEMM
// done 13x with V_WMMA_F32_16X16X4_F32 (f32 needed: third derivatives).
// Jets ping-pong through LDS between layers; tanh jet-composition is pointwise
// in the WMMA C layout so it needs no shuffles.
// tanh is computed branchlessly as 1 - 2/(exp2(2*log2e*z)+1): one v_exp_f32
// (TRANS, co-executes with VALU) + one v_rcp_f32, correct saturation at +-inf,
// no EXEC-divergent libm slow path.

typedef __attribute__((ext_vector_type(2))) float v2f;
typedef __attribute__((ext_vector_type(8))) float v8f;

#define NCOMP 13          // jet components: v,x,y,t,xx,xy,yy,xt,yt,xxx,xxy,xyy,yyy
#define PTS   16          // points per block = WMMA M tile
#define STRIDE 20         // padded neuron stride in LDS (max width)

__device__ __forceinline__ float fast_tanh(float z) {
  // 2*log2(e) = 2.8853900817779268
  float e = __builtin_amdgcn_exp2f(z * 2.8853901f);
  return 1.0f - 2.0f * __builtin_amdgcn_rcpf(e + 1.0f);
}

// ---------------- Kernel 1: min/max of X[:,0] ----------------
__global__ __launch_bounds__(256)
void pinn_minmax(const float* __restrict__ X, int N, float* __restrict__ mm) {
  __shared__ float smin[256], smax[256];
  float lo = 3.402823466e38f, hi = -3.402823466e38f;
  for (int i = threadIdx.x; i < N; i += 256) {
    float v = X[i * 3];
    lo = fminf(lo, v);
    hi = fmaxf(hi, v);
  }
  smin[threadIdx.x] = lo;
  smax[threadIdx.x] = hi;
  __syncthreads();
  for (int s = 128; s > 0; s >>= 1) {
    if ((int)threadIdx.x < s) {
      smin[threadIdx.x] = fminf(smin[threadIdx.x], smin[threadIdx.x + s]);
      smax[threadIdx.x] = fmaxf(smax[threadIdx.x], smax[threadIdx.x + s]);
    }
    __syncthreads();
  }
  if (threadIdx.x == 0) { mm[0] = smin[0]; mm[1] = smax[0]; }
}

// ---------------- Kernel 2: jet propagation + residual ----------------
__global__ __launch_bounds__(32)
void pinn_main(const float* __restrict__ X,
               const float* __restrict__ W1, const float* __restrict__ B1,
               const float* __restrict__ W2, const float* __restrict__ B2,
               const float* __restrict__ W3, const float* __restrict__ B3,
               const float* __restrict__ W4, const float* __restrict__ B4,
               const float* __restrict__ W5, const float* __restrict__ B5,
               const float* __restrict__ W6, const float* __restrict__ B6,
               const float* __restrict__ W7, const float* __restrict__ B7,
               const float* __restrict__ W8, const float* __restrict__ B8,
               const float* __restrict__ lam1p, const float* __restrict__ lam2p,
               const float* __restrict__ mm,
               float* __restrict__ out, int N) {
  __shared__ __align__(16) float jets[2][NCOMP][PTS][STRIDE];

  const int lane = threadIdx.x;          // wave32: one wave per block
  const int half = lane >> 4;            // 0: lanes 0-15, 1: lanes 16-31
  const int l16  = lane & 15;
  const int base_pt = blockIdx.x * PTS;

  // Zero both jet buffers (padding K slots must read as 0).
  for (int i = lane; i < 2 * NCOMP * PTS * STRIDE; i += 32)
    ((float*)jets)[i] = 0.0f;
  __syncthreads();

  const float lb = mm[0], ub = mm[1];
  const float kk = 2.0f / (ub - lb);
  const float cc = -2.0f * lb / (ub - lb) - 1.0f;

  // Input jets (buffer 0). h_j = kk*x_j + cc; dh_j/dx_a = kk * delta(j,a).
  if (half == 0) {
    const int p = base_pt + l16;
    if (p < N) {
      jets[0][0][l16][0] = kk * X[p * 3 + 0] + cc;
      jets[0][0][l16][1] = kk * X[p * 3 + 1] + cc;
      jets[0][0][l16][2] = kk * X[p * 3 + 2] + cc;
      jets[0][1][l16][0] = kk;   // d/dx
      jets[0][2][l16][1] = kk;   // d/dy
      jets[0][3][l16][2] = kk;   // d/dt
    }
  }
  __syncthreads();

  const float* Ws[8] = {W1, W2, W3, W4, W5, W6, W7, W8};
  const float* Bs[8] = {B1, B2, B3, B4, B5, B6, B7, B8};
  const int dins[8]  = {3, 3, 20, 20, 20, 20, 20, 20};
  const int douts[8] = {3, 20, 20, 20, 20, 20, 20, 2};

  int bufin = 0;
#pragma unroll
  for (int L = 0; L < 8; ++L) {
    const int din = dins[L], dout = douts[L];
    const int KC = (din + 3) >> 2;        // K-chunks of 4
    const int NT = (dout + 15) >> 4;      // N-tiles of 16
    const float* __restrict__ W = Ws[L];
    const float* __restrict__ bias = Bs[L];
    const int bufout = bufin ^ 1;

#pragma unroll
    for (int nt = 0; nt < NT; ++nt) {
      const int ncol = nt * 16 + l16;     // this lane's output column
      v8f acc[NCOMP];
#pragma unroll
      for (int q = 0; q < NCOMP; ++q)
        acc[q] = (v8f){0.f, 0.f, 0.f, 0.f, 0.f, 0.f, 0.f, 0.f};

#pragma unroll
      for (int kc = 0; kc < KC; ++kc) {
        // B operand (4x16 f32, K x N): VGPR0 -> K = kc*4 + 2*half,
        // VGPR1 -> K+1 (mirrors the documented 16x4 A layout). Zero-padded.
        const int K0 = kc * 4 + 2 * half;
        v2f bmat;
        bmat.x = (K0 < din && ncol < dout)     ? W[K0 * dout + ncol]       : 0.f;
        bmat.y = (K0 + 1 < din && ncol < dout) ? W[(K0 + 1) * dout + ncol] : 0.f;
#pragma unroll
        for (int q = 0; q < NCOMP; ++q) {
          // A operand (16x4 f32): lane<16 holds K=kc*4+{0,1}, lane>=16 K=+{2,3}
          const float* src = &jets[bufin][q][l16][kc * 4 + 2 * half];
          v2f a;
          a.x = src[0];
          a.y = src[1];
          acc[q] = __builtin_amdgcn_wmma_f32_16x16x4_f32(
              false, a, false, bmat, (short)0, acc[q], false, false);
        }
      }

      // Bias (value component only) + tanh jet composition, pointwise in
      // C layout: VGPR r <-> row M = r + 8*half, lane <-> column ncol.
      const float bv = (ncol < dout) ? bias[ncol] : 0.f;
#pragma unroll
      for (int r = 0; r < 8; ++r) {
        const float z    = acc[0][r] + bv;
        const float zx   = acc[1][r],  zy   = acc[2][r],  zt   = acc[3][r];
        const float zxx  = acc[4][r],  zxy  = acc[5][r],  zyy  = acc[6][r];
        const float zxt  = acc[7][r],  zyt  = acc[8][r];
        const float zxxx = acc[9][r],  zxxy = acc[10][r];
        const float zxyy = acc[11][r], zyyy = acc[12][r];
        const float s  = fast_tanh(z);
        const float d1 = 1.0f - s * s;                 // s'
        const float d2 = -2.0f * s * d1;               // s''
        const float d3 = d1 * (6.0f * s * s - 2.0f);   // s'''
        acc[0][r]  = s;
        acc[1][r]  = d1 * zx;
        acc[2][r]  = d1 * zy;
        acc[3][r]  = d1 * zt;
        acc[4][r]  = d2 * zx * zx + d1 * zxx;
        acc[5][r]  = d2 * zx * zy + d1 * zxy;
        acc[6][r]  = d2 * zy * zy + d1 * zyy;
        acc[7][r]  = d2 * zx * zt + d1 * zxt;
        acc[8][r]  = d2 * zy * zt + d1 * zyt;
        acc[9][r]  = d3 * zx * zx * zx + 3.0f * d2 * zx * zxx + d1 * zxxx;
        acc[10][r] = d3 * zx * zx * zy + d2 * (2.0f * zx * zxy + zy * zxx) + d1 * zxxy;
        acc[11][r] = d3 * zx * zy * zy + d2 * (2.0f * zy * zxy + zx * zyy) + d1 * zxyy;
        acc[12][r] = d3 * zy * zy * zy + 3.0f * d2 * zy * zyy + d1 * zyyy;
      }

      // Store C layout -> LDS [comp][M][ncol].
      if (ncol < dout) {
#pragma unroll
        for (int q = 0; q < NCOMP; ++q)
#pragma unroll
          for (int r = 0; r < 8; ++r)
            jets[bufout][q][r + 8 * half][ncol] = acc[q][r];
      }
    }
    __syncthreads();
    bufin ^= 1;
  }

  // Epilogue: neuron 0 = psi, neuron 1 = p (both tanh outputs).
  if (half == 0) {
    const int m = l16;
    const int p = base_pt + m;
    if (p < N) {
      const float lam1 = lam1p[0], lam2 = lam2p[0];
      const float u    =  jets[bufin][2][m][0];    // psi_y
      const float v    = -jets[bufin][1][m][0];    // -psi_x
      const float pval =  jets[bufin][0][m][1];
      const float p_x  =  jets[bufin][1][m][1];
      const float p_y  =  jets[bufin][2][m][1];
      const float u_t  =  jets[bufin][8][m][0];    // psi_yt
      const float u_x  =  jets[bufin][5][m][0];    // psi_xy
      const float u_y  =  jets[bufin][6][m][0];    // psi_yy
      const float v_t  = -jets[bufin][7][m][0];    // -psi_xt
      const float v_x  = -jets[bufin][4][m][0];    // -psi_xx
      const float v_y  = -jets[bufin][5][m][0];    // -psi_xy
      const float u_xx =  jets[bufin][10][m][0];   // psi_xxy
      const float u_yy =  jets[bufin][12][m][0];   // psi_yyy
      const float v_xx = -jets[bufin][9][m][0];    // -psi_xxx
      const float v_yy = -jets[bufin][11][m][0];   // -psi_xyy
      const float f_u = u_t + lam1 * (u * u_x + v * u_y) + p_x - lam2 * (u_xx + u_yy);
      const float f_v = v_t + lam1 * (u * v_x + v * v_y) + p_y - lam2 * (v_xx + v_yy);
      out[p]         = u;
      out[N + p]     = v;
      out[2 * N + p] = pval;
      out[3 * N + p] = f_u;
      out[4 * N + p] = f_v;
    }
  }
}

extern "C" void kernel_launch(void* const* d_in, const int* in_sizes, int n_in,
                              void* d_out, int out_size, void* d_ws, size_t ws_size,
                              hipStream_t stream) {
  const float* X = (const float*)d_in[0];
  const int N = in_sizes[0] / 3;
  const float* W[8];
  const float* B[8];
  for (int i = 0; i < 8; ++i) {
    W[i] = (const float*)d_in[1 + 2 * i];
    B[i] = (const float*)d_in[2 + 2 * i];
  }
  const float* lam1 = (const float*)d_in[17];
  const float* lam2 = (const float*)d_in[18];
  float* mm = (float*)d_ws;   // [0]=lb, [1]=ub, rewritten every call
  float* out = (float*)d_out;

  pinn_minmax<<<1, 256, 0, stream>>>(X, N, mm);

  const int nblocks = (N + PTS - 1) / PTS;
  pinn_main<<<nblocks, 32, 0, stream>>>(
      X, W[0], B[0], W[1], B[1], W[2], B[2], W[3], B[3],
      W[4], B[4], W[5], B[5], W[6], B[6], W[7], B[7],
      lam1, lam2, mm, out, N);
}